// EGGNEncoder_57801669870185
// MI455X (gfx1250) — compile-verified
//
#include <hip/hip_runtime.h>
#include <math.h>

// EGNN encoder for gfx1250 (MI455X), fp32 end-to-end via V_WMMA_F32_16X16X4_F32.
//
// d_in flattening assumption (top-level dict order, nested params in JAX
// pytree sorted-key order):
//   0 atomic_numbers[int32 N], 1 pos[f32 N*3], 2 edge_index[int32 2*E],
//   3 edge_attr[f32 E], 4 embedding[f32 100*128],
//   5.. per layer (14 each): phi_e1.W,.b, phi_e2.W,.b, phi_h1.W,.b,
//        phi_h2.W,.b, phi_inf.W,.b, phi_x1.W,.b, phi_x2.W,.b
//
// WMMA f32 16x16x4 operand layout (ISA 7.12.2): for lane L, m=L&15, hi=L>>4,
//   A: a = { A[m][4*k4+2*hi], A[m][4*k4+2*hi+1] }       (one float2 load)
//   B: b = { B[4*k4+2*hi][m+16n], B[4*k4+2*hi+1][m+16n] }
//   C/D: acc[n][v] = C[v+8*hi][16n+m]
// Weights are pre-swizzled so B pairs are contiguous float2s:
//   Bsw[kk*128+n] = { W[2kk][n], W[2kk+1][n] }  (zero-padded past K)

typedef __attribute__((ext_vector_type(2))) float v2f;
typedef __attribute__((ext_vector_type(8))) float v8f;

#define WVPB 4  // waves per block (128 threads, wave32)

__device__ __forceinline__ v2f ld2(const float* p) { return *(const v2f*)p; }

__device__ __forceinline__ v8f wmma4(v2f a, v2f b, v8f c) {
  return __builtin_amdgcn_wmma_f32_16x16x4_f32(false, a, false, b, (short)0, c,
                                               false, false);
}

__device__ __forceinline__ float silu_f(float t) {
  return t / (1.0f + __expf(-t));
}

// ---------------------------------------------------------------- utilities
__global__ void k_embed(const int* __restrict__ an,
                        const float* __restrict__ emb,
                        float* __restrict__ h, int total) {
  int i = blockIdx.x * blockDim.x + threadIdx.x;
  if (i < total) h[i] = emb[(size_t)an[i >> 7] * 128 + (i & 127)];
}

__global__ void k_zero(float* __restrict__ p, int n) {
  int i = blockIdx.x * blockDim.x + threadIdx.x;
  if (i < n) p[i] = 0.0f;
}

__global__ void k_copy(const float* __restrict__ a, float* __restrict__ b,
                       int n) {
  int i = blockIdx.x * blockDim.x + threadIdx.x;
  if (i < n) b[i] = a[i];
}

// Bsw[kk*128+n] = {W[2kk][n], W[(2kk+1)][n]}, zero padded for rows >= K.
__global__ void k_swizzle(const float* __restrict__ W, float* __restrict__ o,
                          int K, int Kp) {
  int i = blockIdx.x * blockDim.x + threadIdx.x;
  int total = (Kp / 2) * 128;
  if (i >= total) return;
  int kk = i >> 7, n = i & 127;
  int k0 = 2 * kk;
  o[2 * i + 0] = (k0 < K) ? W[(size_t)k0 * 128 + n] : 0.0f;
  o[2 * i + 1] = (k0 + 1 < K) ? W[(size_t)(k0 + 1) * 128 + n] : 0.0f;
}

// ---------------------------------------------------------------- edge phase
// One wave = one 16-edge tile. Requires (E/16) % WVPB == 0 (800000/16=50000,
// 50000/4=12500 blocks, exact).
__global__ __launch_bounds__(32 * WVPB) void k_edge(
    const float* __restrict__ h, const float* __restrict__ x,
    const int* __restrict__ ei, const float* __restrict__ ea,
    const float* __restrict__ e1sw, const float* __restrict__ be1,
    const float* __restrict__ e2sw, const float* __restrict__ be2,
    const float* __restrict__ x1sw, const float* __restrict__ bx1,
    const float* __restrict__ wx2, const float* __restrict__ bx2,
    const float* __restrict__ winf, const float* __restrict__ binf,
    float* __restrict__ mi, float* __restrict__ xo, int nE) {
  __shared__ float S[WVPB][16 * 132];
  __shared__ int sSrc[WVPB][16], sDst[WVPB][16];
  __shared__ float sR[WVPB][16], sEA[WVPB][16];
  __shared__ float sXD[WVPB][48], sG[WVPB][16], sWij[WVPB][16];

  const int lane = threadIdx.x & 31;
  const int wv = threadIdx.x >> 5;
  const int m = lane & 15, hi = lane >> 4;
  const int e0 = (blockIdx.x * WVPB + wv) * 16;

  if (lane < 16) {
    int e = e0 + lane;
    int s = ei[e], d = ei[nE + e];
    float dx = x[s * 3 + 0] - x[d * 3 + 0];
    float dy = x[s * 3 + 1] - x[d * 3 + 1];
    float dz = x[s * 3 + 2] - x[d * 3 + 2];
    sSrc[wv][lane] = s;
    sDst[wv][lane] = d;
    sXD[wv][lane * 3 + 0] = dx;
    sXD[wv][lane * 3 + 1] = dy;
    sXD[wv][lane * 3 + 2] = dz;
    sR[wv][lane] = sqrtf(dx * dx + dy * dy + dz * dz);
    sEA[wv][lane] = ea[e];
  }
  __syncthreads();

  const float* hD = h + (size_t)sDst[wv][m] * 128;  // h_i = h[dst]
  const float* hS = h + (size_t)sSrc[wv][m] * 128;  // h_j = h[src]
  const float rv = sR[wv][m], eav = sEA[wv][m];
  float* Sw = &S[wv][0];

  // ---- GEMM1: silu(phi_e1([h_i|h_j|r|ea])) ; K padded 258 -> 260
  v8f acc[8];
#pragma unroll
  for (int n = 0; n < 8; n++)
#pragma unroll
    for (int v = 0; v < 8; v++) acc[n][v] = 0.0f;

  for (int k4 = 0; k4 < 65; k4++) {
    int c = k4 * 4 + hi * 2;
    v2f a;
    if (c < 128)
      a = ld2(hD + c);
    else if (c < 256)
      a = ld2(hS + (c - 128));
    else if (c == 256) {
      a.x = rv;
      a.y = eav;
    } else {
      a.x = 0.0f;
      a.y = 0.0f;
    }
    const float* bp = e1sw + ((size_t)(k4 * 2 + hi) * 128 + m) * 2;
#pragma unroll
    for (int n = 0; n < 8; n++) acc[n] = wmma4(a, ld2(bp + n * 32), acc[n]);
  }
#pragma unroll
  for (int n = 0; n < 8; n++) {
    float bias = be1[n * 16 + m];
#pragma unroll
    for (int v = 0; v < 8; v++)
      Sw[(v + 8 * hi) * 132 + n * 16 + m] = silu_f(acc[n][v] + bias);
  }
  __syncthreads();

  // ---- GEMM2: m_ij = silu(phi_e2(t1)), K=128
  v8f ac2[8];
#pragma unroll
  for (int n = 0; n < 8; n++)
#pragma unroll
    for (int v = 0; v < 8; v++) ac2[n][v] = 0.0f;
  for (int k4 = 0; k4 < 32; k4++) {
    int c = k4 * 4 + hi * 2;
    v2f a = ld2(Sw + m * 132 + c);
    const float* bp = e2sw + ((size_t)(k4 * 2 + hi) * 128 + m) * 2;
#pragma unroll
    for (int n = 0; n < 8; n++) ac2[n] = wmma4(a, ld2(bp + n * 32), ac2[n]);
  }
#pragma unroll
  for (int n = 0; n < 8; n++) {
    float bias = be2[n * 16 + m];
#pragma unroll
    for (int v = 0; v < 8; v++)
      Sw[(v + 8 * hi) * 132 + n * 16 + m] = silu_f(ac2[n][v] + bias);
  }
  __syncthreads();

  // ---- gate = sigmoid(phi_inf(m_ij)) : 2 lanes per edge, 64 cols each
  {
    float part = 0.0f;
    const float* wrow = winf + hi * 64;
    const float* srow = Sw + m * 132 + hi * 64;
    for (int c = 0; c < 64; c++) part += srow[c] * wrow[c];
    part += __shfl_xor(part, 16, 32);
    float gate = 1.0f / (1.0f + __expf(-(part + binf[0])));
    if (lane < 16) sG[wv][lane] = gate;
  }
  __syncthreads();

  // ---- m_i[dst] += m_ij * gate (atomic segment-sum, lane-consecutive addrs)
  for (int e = 0; e < 16; e++) {
    float g = sG[wv][e];
    float* p = mi + (size_t)sDst[wv][e] * 128 + lane * 4;
    const float* sp = Sw + e * 132 + lane * 4;
#pragma unroll
    for (int j = 0; j < 4; j++) atomicAdd(p + j, sp[j] * g);
  }

  // ---- GEMM3: t2 = silu(phi_x1(m_ij)); W_ij = phi_x2(t2) fused in regs
  v8f ac3[8];
#pragma unroll
  for (int n = 0; n < 8; n++)
#pragma unroll
    for (int v = 0; v < 8; v++) ac3[n][v] = 0.0f;
  for (int k4 = 0; k4 < 32; k4++) {
    int c = k4 * 4 + hi * 2;
    v2f a = ld2(Sw + m * 132 + c);
    const float* bp = x1sw + ((size_t)(k4 * 2 + hi) * 128 + m) * 2;
#pragma unroll
    for (int n = 0; n < 8; n++) ac3[n] = wmma4(a, ld2(bp + n * 32), ac3[n]);
  }
  {
    float p[8];
#pragma unroll
    for (int v = 0; v < 8; v++) p[v] = 0.0f;
#pragma unroll
    for (int n = 0; n < 8; n++) {
      float bias = bx1[n * 16 + m];
      float w2 = wx2[n * 16 + m];
#pragma unroll
      for (int v = 0; v < 8; v++) p[v] += silu_f(ac3[n][v] + bias) * w2;
    }
#pragma unroll
    for (int mask = 1; mask < 16; mask <<= 1)
#pragma unroll
      for (int v = 0; v < 8; v++) p[v] += __shfl_xor(p[v], mask, 32);
    if (m == 0) {
#pragma unroll
      for (int v = 0; v < 8; v++) sWij[wv][8 * hi + v] = p[v] + bx2[0];
    }
  }
  __syncthreads();

  // ---- x_out[dst] += W_ij * x_diff
  if (lane < 16) {
    float w = sWij[wv][lane];
    size_t d3 = (size_t)sDst[wv][lane] * 3;
    atomicAdd(xo + d3 + 0, w * sXD[wv][lane * 3 + 0]);
    atomicAdd(xo + d3 + 1, w * sXD[wv][lane * 3 + 1]);
    atomicAdd(xo + d3 + 2, w * sXD[wv][lane * 3 + 2]);
  }
}

// ---------------------------------------------------------------- node phase
// h_out = h + phi_h2(silu(phi_h1([h | m_i]))). Out-of-range waves clamp to the
// last tile (duplicate identical plain stores: deterministic, no atomics).
__global__ __launch_bounds__(32 * WVPB) void k_node(
    const float* __restrict__ h, const float* __restrict__ mi,
    const float* __restrict__ h1sw, const float* __restrict__ bh1,
    const float* __restrict__ h2sw, const float* __restrict__ bh2,
    float* __restrict__ ho, int nTiles) {
  __shared__ float S[WVPB][16 * 132];
  const int lane = threadIdx.x & 31;
  const int wv = threadIdx.x >> 5;
  const int m = lane & 15, hi = lane >> 4;
  int tile = blockIdx.x * WVPB + wv;
  if (tile >= nTiles) tile = nTiles - 1;
  const int n0 = tile * 16;

  const float* hR = h + (size_t)(n0 + m) * 128;
  const float* miR = mi + (size_t)(n0 + m) * 128;
  float* Sw = &S[wv][0];

  v8f acc[8];
#pragma unroll
  for (int n = 0; n < 8; n++)
#pragma unroll
    for (int v = 0; v < 8; v++) acc[n][v] = 0.0f;
  for (int k4 = 0; k4 < 64; k4++) {
    int c = k4 * 4 + hi * 2;
    v2f a = (c < 128) ? ld2(hR + c) : ld2(miR + (c - 128));
    const float* bp = h1sw + ((size_t)(k4 * 2 + hi) * 128 + m) * 2;
#pragma unroll
    for (int n = 0; n < 8; n++) acc[n] = wmma4(a, ld2(bp + n * 32), acc[n]);
  }
#pragma unroll
  for (int n = 0; n < 8; n++) {
    float bias = bh1[n * 16 + m];
#pragma unroll
    for (int v = 0; v < 8; v++)
      Sw[(v + 8 * hi) * 132 + n * 16 + m] = silu_f(acc[n][v] + bias);
  }
  __syncthreads();

  v8f ac2[8];
#pragma unroll
  for (int n = 0; n < 8; n++)
#pragma unroll
    for (int v = 0; v < 8; v++) ac2[n][v] = 0.0f;
  for (int k4 = 0; k4 < 32; k4++) {
    int c = k4 * 4 + hi * 2;
    v2f a = ld2(Sw + m * 132 + c);
    const float* bp = h2sw + ((size_t)(k4 * 2 + hi) * 128 + m) * 2;
#pragma unroll
    for (int n = 0; n < 8; n++) ac2[n] = wmma4(a, ld2(bp + n * 32), ac2[n]);
  }
#pragma unroll
  for (int n = 0; n < 8; n++) {
    float bias = bh2[n * 16 + m];
#pragma unroll
    for (int v = 0; v < 8; v++) {
      size_t o = (size_t)(n0 + v + 8 * hi) * 128 + n * 16 + m;
      ho[o] = ac2[n][v] + bias + h[o];
    }
  }
}

// ---------------------------------------------------------------- launcher
extern "C" void kernel_launch(void* const* d_in, const int* in_sizes, int n_in,
                              void* d_out, int out_size, void* d_ws,
                              size_t ws_size, hipStream_t stream) {
  (void)in_sizes; (void)n_in; (void)out_size; (void)ws_size;
  const int N = 50000, E = 800000, H = 128;

  const int* an = (const int*)d_in[0];
  const float* pos = (const float*)d_in[1];
  const int* ei = (const int*)d_in[2];
  const float* ea = (const float*)d_in[3];
  const float* emb = (const float*)d_in[4];

  // workspace carve-up (floats); total ~19.8M floats (~79 MB)
  float* ws = (float*)d_ws;
  float* hA = ws; ws += (size_t)N * H;
  float* hB = ws; ws += (size_t)N * H;
  float* mi = ws; ws += (size_t)N * H;
  float* xA = ws; ws += (size_t)N * 3;
  float* xB = ws; ws += (size_t)N * 3;
  float* e1sw[2]; float* e2sw[2]; float* h1sw[2]; float* h2sw[2]; float* x1sw[2];
  for (int l = 0; l < 2; l++) {
    e1sw[l] = ws; ws += 130 * 128 * 2;  // Kp=260
    e2sw[l] = ws; ws += 64 * 128 * 2;   // Kp=128
    h1sw[l] = ws; ws += 128 * 128 * 2;  // Kp=256
    h2sw[l] = ws; ws += 64 * 128 * 2;
    x1sw[l] = ws; ws += 64 * 128 * 2;
  }

  // pre-swizzle weights into WMMA B layout
  for (int l = 0; l < 2; l++) {
    const int b = 5 + 14 * l;
    const float* We1 = (const float*)d_in[b + 0];
    const float* We2 = (const float*)d_in[b + 2];
    const float* Wh1 = (const float*)d_in[b + 4];
    const float* Wh2 = (const float*)d_in[b + 6];
    const float* Wx1 = (const float*)d_in[b + 10];
    k_swizzle<<<(130 * 128 + 255) / 256, 256, 0, stream>>>(We1, e1sw[l], 258, 260);
    k_swizzle<<<(64 * 128 + 255) / 256, 256, 0, stream>>>(We2, e2sw[l], 128, 128);
    k_swizzle<<<(128 * 128 + 255) / 256, 256, 0, stream>>>(Wh1, h1sw[l], 256, 256);
    k_swizzle<<<(64 * 128 + 255) / 256, 256, 0, stream>>>(Wh2, h2sw[l], 128, 128);
    k_swizzle<<<(64 * 128 + 255) / 256, 256, 0, stream>>>(Wx1, x1sw[l], 128, 128);
  }

  // h0 = embedding[atomic_numbers]
  k_embed<<<((size_t)N * H + 255) / 256, 256, 0, stream>>>(an, emb, hA, N * H);

  const float* xin = pos;
  float* xout = xA;
  float* hin = hA;
  float* hout = hB;
  const int blocksE = (E / 16) / WVPB;                    // 12500 exact
  const int tilesN = N / 16;                              // 3125 exact
  const int blocksN = (tilesN + WVPB - 1) / WVPB;         // 782 (clamped)

  for (int l = 0; l < 2; l++) {
    const int b = 5 + 14 * l;
    const float* be1 = (const float*)d_in[b + 1];
    const float* be2 = (const float*)d_in[b + 3];
    const float* bh1 = (const float*)d_in[b + 5];
    const float* bh2 = (const float*)d_in[b + 7];
    const float* winf = (const float*)d_in[b + 8];
    const float* binf = (const float*)d_in[b + 9];
    const float* bx1 = (const float*)d_in[b + 11];
    const float* wx2 = (const float*)d_in[b + 12];
    const float* bx2 = (const float*)d_in[b + 13];

    k_zero<<<((size_t)N * H + 255) / 256, 256, 0, stream>>>(mi, N * H);
    k_copy<<<(N * 3 + 255) / 256, 256, 0, stream>>>(xin, xout, N * 3);
    k_edge<<<blocksE, 32 * WVPB, 0, stream>>>(
        hin, xin, ei, ea, e1sw[l], be1, e2sw[l], be2, x1sw[l], bx1, wx2, bx2,
        winf, binf, mi, xout, E);
    k_node<<<blocksN, 32 * WVPB, 0, stream>>>(hin, mi, h1sw[l], bh1, h2sw[l],
                                              bh2, hout, tilesN);
    xin = xout;
    xout = xB;
    float* t = hin; hin = hout; hout = t;
  }

  // outputs: h (N*128) then x (N*3), concatenated flat
  float* out = (float*)d_out;
  k_copy<<<((size_t)N * H + 255) / 256, 256, 0, stream>>>(hin, out, N * H);
  k_copy<<<(N * 3 + 255) / 256, 256, 0, stream>>>(xin, out + (size_t)N * H,
                                                  N * 3);
}